// ElasticEnergyLossWithJacobian_29703993819930
// MI455X (gfx1250) — compile-verified
//
#include <hip/hip_runtime.h>

// ---------------------------------------------------------------------------
// ElasticEnergyLossWithJacobian for MI455X (gfx1250, wave32).
//
// Roofline: 170 MB in / 4 B out -> ~7.3us HBM floor @ 23.3 TB/s; ~2.1 GFLOP
// f32 is negligible. Whole tensor fits in the 192 MB L2, so stencil neighbor
// re-reads are L2 hits; coalesced z-row global loads are the optimal path.
//
// CDNA5-specific: V_WMMA_F32_16X16X4_F32 performs the wave reduction.
// Partials are fed as the A-matrix (16x4) against a ones B-matrix, so
// D[m][n] = acc0_m + acc1_m + acc0_{m+16} + acc1_{m+16} for every n.
// Each lane then sums its 8 D VGPRs (rows 0-7 or 8-15) and one
// ds_swizzle SWAPX16 merges the halves -> full 64-partial wave sum with
// 1 WMMA + 7 adds + 1 swizzle (no bpermute chain).
//
// Two-pass reduction through d_ws avoids 2304 same-address atomics (which
// would serialize at L2 and dwarf the 7.3us memory floor).
// ---------------------------------------------------------------------------

typedef __attribute__((ext_vector_type(2))) float v2f;
typedef __attribute__((ext_vector_type(8))) float v8f;

#define DIM    192
#define DIM2   (DIM * DIM)          // 36864
#define DIM3   (DIM * DIM * DIM)    // 7077888
#define NVOX   (2 * DIM3)           // 14155776
#define NPAIRS (DIM * DIM)          // 36864 row-pairs (73728 z-rows / 2)
#define GRID1  2304                 // pass-1 blocks; NPAIRS/GRID1 = 16 pairs each
#define WAVES1 6                    // 192 threads / 32
#define BLK2   256
#define WAVES2 8
#define LAMBDA_LAME 1.0f
#define MU_LAME     0.5f
#define JAC_W       0.1f
#define DELTA       0.01f

__device__ __forceinline__ float robust_f(float x) {
    float ax = fabsf(x);
    return (ax <= DELTA) ? 0.5f * x * x : DELTA * (ax - 0.5f * DELTA);
}

// jnp.gradient along one axis: central diff interior, one-sided at edges.
__device__ __forceinline__ float grad1(const float* __restrict__ p, int c, int stride) {
    float ctr = p[0];
    float lo  = (c > 0)       ? p[-stride] : ctr;
    float hi  = (c < DIM - 1) ? p[ stride] : ctr;
    float s   = (c > 0 && c < DIM - 1) ? 0.5f : 1.0f;
    return (hi - lo) * s;
}

__device__ __forceinline__ float voxel_contrib(const float* __restrict__ df,
                                               int b, int x, int y, int z) {
    const float* pu = df + (size_t)(b * 3 + 0) * DIM3 + (size_t)x * DIM2 + y * DIM + z;
    const float* pv = pu + DIM3;
    const float* pw = pv + DIM3;

    float du_dx = grad1(pu, x, DIM2), du_dy = grad1(pu, y, DIM), du_dz = grad1(pu, z, 1);
    float dv_dx = grad1(pv, x, DIM2), dv_dy = grad1(pv, y, DIM), dv_dz = grad1(pv, z, 1);
    float dw_dx = grad1(pw, x, DIM2), dw_dy = grad1(pw, y, DIM), dw_dz = grad1(pw, z, 1);

    float Exx = du_dx, Eyy = dv_dy, Ezz = dw_dz;
    float Exy = 0.5f * (du_dy + dv_dx);
    float Exz = 0.5f * (du_dz + dw_dx);
    float Eyz = 0.5f * (dv_dz + dw_dy);

    float rxx = robust_f(Exx), ryy = robust_f(Eyy), rzz = robust_f(Ezz);
    float rxy = robust_f(Exy), rxz = robust_f(Exz), ryz = robust_f(Eyz);
    float trl = robust_f(Exx + Eyy + Ezz);

    float energy = 0.5f * LAMBDA_LAME * trl * trl +
                   MU_LAME * (rxx * rxx + ryy * ryy + rzz * rzz +
                              2.0f * (rxy * rxy + rxz * rxz + ryz * ryz));

    float a11 = 1.0f + du_dx, a12 = du_dy,        a13 = du_dz;
    float a21 = dv_dx,        a22 = 1.0f + dv_dy, a23 = dv_dz;
    float a31 = dw_dx,        a32 = dw_dy,        a33 = 1.0f + dw_dz;
    float jac = a11 * (a22 * a33 - a23 * a32)
              - a12 * (a21 * a33 - a23 * a31)
              + a13 * (a21 * a32 - a22 * a31);
    float jac_penalty = fmaxf(-jac, 0.0f);

    return energy + JAC_W * jac_penalty;
}

// One ds_swizzle_b32, group-of-32 xor-16 (SWAPX16): offset = {xor=0x10,or=0,and=0x1f}
__device__ __forceinline__ float swap_half(float v) {
    return __int_as_float(__builtin_amdgcn_ds_swizzle(__float_as_int(v), 0x401F));
}

// Full 32-lane sum of (a0 + a1), returned in every lane.
// WMMA D = A(16x4, partials) x B(4x16, ones) + 0:
//   row m sums lanes {m, m+16}'s two partials; lane L's 8 D-VGPRs hold rows
//   0-7 (L<16) or 8-15 (L>=16). Requires EXEC all-1s: callers are converged.
__device__ __forceinline__ float wave_sum_wmma(float a0, float a1) {
    v2f av;   av.x   = a0;   av.y   = a1;
    v2f ones; ones.x = 1.0f; ones.y = 1.0f;
    v8f cv = {};
    v8f dv = __builtin_amdgcn_wmma_f32_16x16x4_f32(
        /*neg_a=*/false, av, /*neg_b=*/false, ones,
        /*c_mod=*/(short)0, cv, /*reuse_a=*/false, /*reuse_b=*/false);
    float t = ((dv[0] + dv[1]) + (dv[2] + dv[3])) +
              ((dv[4] + dv[5]) + (dv[6] + dv[7]));
    return t + swap_half(t);
}

__global__ void zero_out_kernel(float* out) { out[0] = 0.0f; }

// Pass 1: 192 threads = 6 wave32s; each block accumulates 16 row-pairs
// (uniform trip count -> no divergence; EXEC all-1s at the WMMA).
template <bool USE_WS>
__global__ __launch_bounds__(192)
void ElasticEnergyLossWithJacobian_29703993819930_pass1(const float* __restrict__ df,
                                                        float* __restrict__ partial,
                                                        float* __restrict__ out) {
    const int z = threadIdx.x;  // 0..191, unit-stride axis -> coalesced
    float acc0 = 0.0f, acc1 = 0.0f;

    for (int pair = blockIdx.x; pair < NPAIRS; pair += GRID1) {
        const int row0 = pair * 2;
        int b0 = row0 / DIM2, r0 = row0 - b0 * DIM2;
        int x0 = r0 / DIM,    y0 = r0 - x0 * DIM;
        acc0 += voxel_contrib(df, b0, x0, y0, z);

        const int row1 = row0 + 1;
        int b1 = row1 / DIM2, r1 = row1 - b1 * DIM2;
        int x1 = r1 / DIM,    y1 = r1 - x1 * DIM;
        acc1 += voxel_contrib(df, b1, x1, y1, z);
    }

    float t = wave_sum_wmma(acc0, acc1);

    __shared__ float wsum[WAVES1];
    const int lane = threadIdx.x & 31;
    const int wid  = threadIdx.x >> 5;
    if (lane == 0) wsum[wid] = t;
    __syncthreads();

    if (threadIdx.x == 0) {
        float s = 0.0f;
#pragma unroll
        for (int i = 0; i < WAVES1; ++i) s += wsum[i];
        if (USE_WS) partial[blockIdx.x] = s;                       // no atomics
        else        atomicAdd(out, s * (1.0f / (float)NVOX));      // fallback
    }
}

// Pass 2: single block reduces the 2304 block partials (2304 = 256 * 9).
__global__ __launch_bounds__(BLK2)
void ElasticEnergyLossWithJacobian_29703993819930_pass2(const float* __restrict__ partial,
                                                        float* __restrict__ out) {
    float acc0 = 0.0f, acc1 = 0.0f;
#pragma unroll
    for (int k = 0; k < GRID1 / BLK2; ++k) {   // 9 uniform iterations
        float v = partial[threadIdx.x + k * BLK2];
        if (k & 1) acc1 += v; else acc0 += v;
    }

    float t = wave_sum_wmma(acc0, acc1);

    __shared__ float wsum[WAVES2];
    const int lane = threadIdx.x & 31;
    const int wid  = threadIdx.x >> 5;
    if (lane == 0) wsum[wid] = t;
    __syncthreads();

    if (threadIdx.x == 0) {
        float s = 0.0f;
#pragma unroll
        for (int i = 0; i < WAVES2; ++i) s += wsum[i];
        out[0] = s * (1.0f / (float)NVOX);
    }
}

extern "C" void kernel_launch(void* const* d_in, const int* in_sizes, int n_in,
                              void* d_out, int out_size, void* d_ws, size_t ws_size,
                              hipStream_t stream) {
    (void)in_sizes; (void)n_in; (void)out_size;
    const float* df = (const float*)d_in[0];
    float* out = (float*)d_out;

    if (ws_size >= (size_t)GRID1 * sizeof(float)) {
        float* partial = (float*)d_ws;
        ElasticEnergyLossWithJacobian_29703993819930_pass1<true>
            <<<GRID1, DIM, 0, stream>>>(df, partial, out);
        ElasticEnergyLossWithJacobian_29703993819930_pass2
            <<<1, BLK2, 0, stream>>>(partial, out);
    } else {
        zero_out_kernel<<<1, 1, 0, stream>>>(out);
        ElasticEnergyLossWithJacobian_29703993819930_pass1<false>
            <<<GRID1, DIM, 0, stream>>>(df, nullptr, out);
    }
}